// VGG16RoIHead_74715251081428
// MI455X (gfx1250) — compile-verified
//
#include <hip/hip_runtime.h>
#include <hip/hip_bf16.h>

typedef float v2f __attribute__((ext_vector_type(2)));
typedef float v8f __attribute__((ext_vector_type(8)));

#define RR    256      // rois
#define CC    512      // channels
#define HH    38
#define WW    38
#define PP    7
#define DIN   25088    // 512*7*7
#define DFC   4096
#define KSPLIT 4       // deterministic split-K factor

// ---------------------------------------------------------------------------
// RoI max pooling: feature [N,C,H,W] -> x [R, C*7*7]  (exact reference semantics)
// ---------------------------------------------------------------------------
__global__ __launch_bounds__(256) void roi_pool_kernel(
    const float* __restrict__ feat, const int* __restrict__ rois,
    const int* __restrict__ roi_idx, float* __restrict__ xout) {
  const int r   = blockIdx.y;
  const int idx = blockIdx.x * blockDim.x + threadIdx.x;   // over C*49
  if (idx >= CC * PP * PP) return;
  const int c  = idx / (PP * PP);
  const int s  = idx % (PP * PP);
  const int ph = s / PP;
  const int pw = s % PP;

  const int n = roi_idx[r];
  const float x1 = (float)rois[r * 4 + 0];
  const float y1 = (float)rois[r * 4 + 1];
  const float x2 = (float)rois[r * 4 + 2];
  const float y2 = (float)rois[r * 4 + 3];
  // jnp.round == round-to-nearest-even == rintf in default device rounding mode
  const int sw = (int)rintf(x1 * 0.0625f);
  const int ew = (int)rintf(x2 * 0.0625f);
  const int sh = (int)rintf(y1 * 0.0625f);
  const int eh = (int)rintf(y2 * 0.0625f);
  const float roi_w = fmaxf((float)(ew - sw + 1), 1.0f);
  const float roi_h = fmaxf((float)(eh - sh + 1), 1.0f);
  const float bw = roi_w / (float)PP;
  const float bh = roi_h / (float)PP;

  int ws = min(max((int)floorf((float)pw * bw) + sw, 0), WW);
  int we = min(max((int)ceilf((float)(pw + 1) * bw) + sw, 0), WW);
  int hs = min(max((int)floorf((float)ph * bh) + sh, 0), HH);
  int he = min(max((int)ceilf((float)(ph + 1) * bh) + sh, 0), HH);
  we = min(we, ws + 8);    // reference caps the window at K=8
  he = min(he, hs + 8);

  const float* fp = feat + (((size_t)n * CC) + c) * (HH * WW);
  float m = -__builtin_inff();
  for (int y = hs; y < he; ++y)
    for (int x = ws; x < we; ++x)
      m = fmaxf(m, fp[y * WW + x]);

  xout[(size_t)r * DIN + idx] = (m == -__builtin_inff()) ? 0.0f : m;
}

// ---------------------------------------------------------------------------
// FP32 WMMA GEMM (split-K partial, double-buffered LDS weight staging):
//   Cpart[ks][256,N] = A[256, k0..k0+Ks) @ W[k0..k0+Ks, N-tile)
// block = 512 threads (16 waves); wave w owns rows [16w,16w+16) and FOUR
// 16-col N-tiles (64 cols per block) -> 4 independent accumulator chains and
// 4x A-fragment reuse. K chunked by 128. W tile (128x64) staged through
// registers into one of two LDS buffers while the other is being consumed;
// a single barrier per chunk separates the store of buffer b from the
// previous chunk's reads of buffer b. LDS layout [j][col][lo][c] makes each
// B fragment one conflict-free ds_load_b64.
// ---------------------------------------------------------------------------
__global__ __launch_bounds__(512) void gemm_wmma_f32(
    const float* __restrict__ A, const float* __restrict__ Wt,
    float* __restrict__ Cpart, int K, int N) {
  __shared__ float wl[2][8192];       // 2 x (128 x 64) swizzled, 64 KB

  const int tid  = threadIdx.x;
  const int wave = tid >> 5;
  const int lane = tid & 31;
  const int hi   = lane >> 4;         // half-wave select (K pairing)
  const int nn   = lane & 15;         // column-within-tile / A row-within-tile
  const int m0   = wave * 16;
  const int nb   = blockIdx.x * 64;   // N base of this block

  const int Ks = K / KSPLIT;          // K-slice length (multiple of 128)
  const int k0 = blockIdx.y * Ks;

  v8f acc0 = {0.f,0.f,0.f,0.f,0.f,0.f,0.f,0.f};
  v8f acc1 = acc0, acc2 = acc0, acc3 = acc0;

  const int    m    = m0 + nn;                            // A row for this lane
  const float* arow = A + (size_t)m * K + k0 + 2 * hi;    // fragment K offset
  const int    srow = tid >> 6;                           // staging row 0..7
  const int    scol = tid & 63;                           // staging col 0..63

  // W staging helpers (16 elements per thread per 128x64 tile)
  float regs[16];
  auto stage_load = [&](int kb) {
#pragma unroll
    for (int i = 0; i < 16; ++i) {
      const int rr = srow + 8 * i;                        // 0..127
      regs[i] = Wt[(size_t)(k0 + kb + rr) * N + nb + scol];
    }
  };
  auto stage_store = [&](int buf) {
#pragma unroll
    for (int i = 0; i < 16; ++i) {
      const int rr = srow + 8 * i;
      // swizzle: idx = [j][col][lo][c], j=rr>>2, lo=(rr>>1)&1, c=rr&1
      wl[buf][(rr >> 2) * 256 + scol * 4 + ((rr >> 1) & 1) * 2 + (rr & 1)] =
          regs[i];
    }
  };

  const int nchunks = Ks >> 7;        // Ks / 128
  stage_load(0);
  stage_store(0);
  __syncthreads();

  for (int ch = 0; ch < nchunks; ++ch) {
    const int kb = ch << 7;
    if (ch + 1 < nchunks) stage_load(kb + 128);           // prefetch next tile

    const float* ap = arow + kb;
    const float* bbase = &wl[ch & 1][nn * 4 + hi * 2];
#pragma unroll 4
    for (int j = 0; j < 32; ++j) {
      // A 16x4 fragment: lanes 0-15 -> K0,K1 ; lanes 16-31 -> K2,K3
      v2f a = *(const v2f*)(ap + 4 * j);
      const float* bp = bbase + j * 256;
      v2f b0 = *(const v2f*)(bp);           // N-tile 0
      v2f b1 = *(const v2f*)(bp + 64);      // N-tile 1 (16 cols * 4)
      v2f b2 = *(const v2f*)(bp + 128);     // N-tile 2
      v2f b3 = *(const v2f*)(bp + 192);     // N-tile 3
      acc0 = __builtin_amdgcn_wmma_f32_16x16x4_f32(false, a, false, b0,
                                                   (short)0, acc0, false, false);
      acc1 = __builtin_amdgcn_wmma_f32_16x16x4_f32(false, a, false, b1,
                                                   (short)0, acc1, false, false);
      acc2 = __builtin_amdgcn_wmma_f32_16x16x4_f32(false, a, false, b2,
                                                   (short)0, acc2, false, false);
      acc3 = __builtin_amdgcn_wmma_f32_16x16x4_f32(false, a, false, b3,
                                                   (short)0, acc3, false, false);
    }

    if (ch + 1 < nchunks) stage_store((ch + 1) & 1);      // fill other buffer
    __syncthreads();   // separates store(buf) from prior reads of buf
  }

  // C layout: VGPR j holds (row = j + 8*hi, col = nn) of each 16x16 tile
  float* cp = Cpart + (size_t)blockIdx.y * RR * N;
#pragma unroll
  for (int j = 0; j < 8; ++j) {
    const size_t row = (size_t)(m0 + j + 8 * hi) * N + nb + nn;
    cp[row +  0] = acc0[j];
    cp[row + 16] = acc1[j];
    cp[row + 32] = acc2[j];
    cp[row + 48] = acc3[j];
  }
}

// ---------------------------------------------------------------------------
// Deterministic split-K reduction + bias + optional ReLU
// ---------------------------------------------------------------------------
__global__ __launch_bounds__(256) void reduce_k_kernel(
    const float* __restrict__ Cpart, const float* __restrict__ bias,
    float* __restrict__ C, int N, int do_relu) {
  const int i = blockIdx.x * blockDim.x + threadIdx.x;
  if (i >= RR * N) return;
  const int n = i % N;
  float v = bias[n];
#pragma unroll
  for (int s = 0; s < KSPLIT; ++s) v += Cpart[(size_t)s * RR * N + i];
  if (do_relu) v = fmaxf(v, 0.0f);
  C[i] = v;
}

// ---------------------------------------------------------------------------
// Head: loc = fc7 @ W_loc + b_loc ; scores = softmax(fc7 @ W_score + b_score)
// one block (16 waves) per RoI; wave w computes output column w (15 used).
// ---------------------------------------------------------------------------
__global__ __launch_bounds__(512) void head_kernel(
    const float* __restrict__ fc7,
    const float* __restrict__ Wloc, const float* __restrict__ bloc,
    const float* __restrict__ Wsc,  const float* __restrict__ bsc,
    float* __restrict__ out) {
  const int r    = blockIdx.x;
  const int wave = threadIdx.x >> 5;
  const int lane = threadIdx.x & 31;
  __shared__ float res[16];

  const float* a = fc7 + (size_t)r * DFC;
  if (wave < 15) {
    const float* Wp;
    int col, nc;
    float bv;
    if (wave < 12) { Wp = Wloc; col = wave;      nc = 12; bv = bloc[col]; }
    else           { Wp = Wsc;  col = wave - 12; nc = 3;  bv = bsc[col]; }
    float s = 0.0f;
    for (int k = lane; k < DFC; k += 32) s += a[k] * Wp[(size_t)k * nc + col];
#pragma unroll
    for (int off = 16; off > 0; off >>= 1) s += __shfl_down(s, off, 32);
    if (lane == 0) res[wave] = s + bv;
  }
  __syncthreads();

  const int t = threadIdx.x;
  if (t < 12) out[(size_t)r * 12 + t] = res[t];
  if (t == 0) {
    const float z0 = res[12], z1 = res[13], z2 = res[14];
    const float mx = fmaxf(z0, fmaxf(z1, z2));
    const float e0 = __expf(z0 - mx), e1 = __expf(z1 - mx), e2 = __expf(z2 - mx);
    const float inv = 1.0f / (e0 + e1 + e2);
    float* sc = out + (size_t)RR * 12;
    sc[r * 3 + 0] = e0 * inv;
    sc[r * 3 + 1] = e1 * inv;
    sc[r * 3 + 2] = e2 * inv;
  }
}

// ---------------------------------------------------------------------------
extern "C" void kernel_launch(void* const* d_in, const int* in_sizes, int n_in,
                              void* d_out, int out_size, void* d_ws, size_t ws_size,
                              hipStream_t stream) {
  const float* feat    = (const float*)d_in[0];
  const int*   rois    = (const int*)  d_in[1];
  const int*   roi_idx = (const int*)  d_in[2];
  const float* W_fc6   = (const float*)d_in[3];
  const float* b_fc6   = (const float*)d_in[4];
  const float* W_fc7   = (const float*)d_in[5];
  const float* b_fc7   = (const float*)d_in[6];
  const float* W_loc   = (const float*)d_in[7];
  const float* b_loc   = (const float*)d_in[8];
  const float* W_sc    = (const float*)d_in[9];
  const float* b_sc    = (const float*)d_in[10];
  float* out = (float*)d_out;

  float* xbuf   = (float*)d_ws;                        // 256 x 25088 (25.7 MB)
  float* fc6buf = xbuf   + (size_t)RR * DIN;           // 256 x 4096  (4 MB)
  float* fc7buf = fc6buf + (size_t)RR * DFC;           // 256 x 4096  (4 MB)
  float* pbuf   = fc7buf + (size_t)RR * DFC;           // 4 x 256 x 4096 (16 MB)

  // 1) RoI max pool -> x
  {
    dim3 grid((CC * PP * PP + 255) / 256, RR);
    roi_pool_kernel<<<grid, 256, 0, stream>>>(feat, rois, roi_idx, xbuf);
  }
  // 2) fc6 = relu(x @ W_fc6 + b)   K=25088, N=4096, split-K=4
  {
    dim3 grid(DFC / 64, KSPLIT);
    gemm_wmma_f32<<<grid, 512, 0, stream>>>(xbuf, W_fc6, pbuf, DIN, DFC);
    reduce_k_kernel<<<(RR * DFC + 255) / 256, 256, 0, stream>>>(
        pbuf, b_fc6, fc6buf, DFC, 1);
  }
  // 3) fc7 = relu(fc6 @ W_fc7 + b) K=4096, N=4096, split-K=4
  {
    dim3 grid(DFC / 64, KSPLIT);
    gemm_wmma_f32<<<grid, 512, 0, stream>>>(fc6buf, W_fc7, pbuf, DFC, DFC);
    reduce_k_kernel<<<(RR * DFC + 255) / 256, 256, 0, stream>>>(
        pbuf, b_fc7, fc7buf, DFC, 1);
  }
  // 4) loc + softmax(scores)
  head_kernel<<<RR, 512, 0, stream>>>(fc7buf, W_loc, b_loc, W_sc, b_sc, out);
}